// GraphAttentionLayer_38611755991228
// MI455X (gfx1250) — compile-verified
//
#include <hip/hip_runtime.h>
#include <hip/hip_bf16.h>
#include <math.h>

#define N_IN_F   128
#define N_OUT_C  128   // HEADS * OUT_F
#define N_HEADS  4
#define N_OUT_F  32
#define NEG_SLOPE 0.2f
#define GAT_EPS   1e-8f

typedef float v2f __attribute__((ext_vector_type(2)));
typedef float v8f __attribute__((ext_vector_type(8)));

// ---------- monotone uint key for float atomic max ----------
__device__ __forceinline__ unsigned fkey(float v) {
    unsigned u = __float_as_uint(v);
    return (u >> 31) ? ~u : (u | 0x80000000u);
}
__device__ __forceinline__ float funkey(unsigned k) {
    unsigned u = (k >> 31) ? (k & 0x7FFFFFFFu) : ~k;
    return __uint_as_float(u);
}

// ---------- 0) zero init: out, att_sum, gmax key ----------
__global__ void gat_zero(float* __restrict__ out, float* __restrict__ att_sum,
                         unsigned* __restrict__ gkey, int nout, int nsum) {
    int i = blockIdx.x * blockDim.x + threadIdx.x;
    if (i < nout) out[i] = 0.0f;
    if (i < nsum) att_sum[i] = 0.0f;
    if (i == 0) *gkey = 0u;   // below fkey(-inf)
}

// ---------- 1) h = x @ W^T  via V_WMMA_F32_16X16X4_F32 ----------
// block = 256 threads = 8 waves; block handles 16 rows; wave w handles cols [16w,16w+16)
__global__ __launch_bounds__(256) void gat_gemm_wmma(const float* __restrict__ x,
                                                     const float* __restrict__ W,
                                                     float* __restrict__ hbuf,
                                                     int nrows) {
    __shared__ float xs[16 * 132];   // pitch 132 (mod 64 = 4) to avoid bank conflicts

    const int tid  = threadIdx.x;
    const int wave = tid >> 5;
    const int lane = tid & 31;
    const int rb   = blockIdx.x * 16;

    // cooperative load of x tile (16 x 128 f32) as float4, padded LDS pitch
    for (int i = tid; i < 16 * 32; i += 256) {
        const int r  = i >> 5;       // row 0..15
        const int c4 = i & 31;       // float4 index within row
        float4 v = make_float4(0.f, 0.f, 0.f, 0.f);
        if (rb + r < nrows)
            v = reinterpret_cast<const float4*>(x + (size_t)(rb + r) * N_IN_F)[c4];
        float* p = &xs[r * 132 + c4 * 4];
        p[0] = v.x; p[1] = v.y; p[2] = v.z; p[3] = v.w;
    }
    __syncthreads();

    const int nb   = wave * 16;      // output-column base for this wave
    const int half = lane >> 4;      // 0: lanes 0-15, 1: lanes 16-31
    const int l15  = lane & 15;

    // A: row M=l15, element j holds K = kbase + j + 2*half  -> consecutive pair
    const float* xrow = &xs[l15 * 132 + 2 * half];
    // B: col N=l15 (B[k][n] = W[n][k]), element j holds K = kbase + j + 2*half
    const float* wrow = W + (size_t)(nb + l15) * N_IN_F + 2 * half;

    v8f acc = {};
    #pragma unroll
    for (int k = 0; k < N_IN_F; k += 4) {
        v2f a = *reinterpret_cast<const v2f*>(xrow + k);   // ds_load_b64
        v2f b = *reinterpret_cast<const v2f*>(wrow + k);   // global_load_b64
        acc = __builtin_amdgcn_wmma_f32_16x16x4_f32(
                  false, a, false, b, (short)0, acc, false, false);
    }

    // D layout: VGPR v -> row M = v + 8*half, col N = l15
    const int row0 = rb + 8 * half;
    float* outp = hbuf + (size_t)row0 * N_OUT_C + nb + l15;
    #pragma unroll
    for (int v = 0; v < 8; ++v) {
        if (row0 + v < nrows) outp[(size_t)v * N_OUT_C] = acc[v];
    }
}

// ---------- 2) per-node attention scalars ----------
__global__ void gat_alpha(const float* __restrict__ hbuf, const float* __restrict__ a,
                          float* __restrict__ alpha_src, float* __restrict__ alpha_dst,
                          int n_nodes) {
    int i = blockIdx.x * blockDim.x + threadIdx.x;  // over N*HEADS
    if (i >= n_nodes * N_HEADS) return;
    const int node = i >> 2;
    const int hd   = i & 3;
    const float* hp = hbuf + (size_t)node * N_OUT_C + hd * N_OUT_F;
    const float* as = a + hd * (2 * N_OUT_F);
    const float* ad = as + N_OUT_F;
    float s1 = 0.f, s2 = 0.f;
    #pragma unroll
    for (int f = 0; f < N_OUT_F; ++f) {
        const float v = hp[f];
        s1 = fmaf(v, as[f], s1);
        s2 = fmaf(v, ad[f], s2);
    }
    alpha_src[i] = s1;
    alpha_dst[i] = s2;
}

// ---------- 3) global max of leakyrelu(att) over (E,H) ----------
__global__ __launch_bounds__(256) void gat_max(const int* __restrict__ src,
                                               const int* __restrict__ dst,
                                               const float* __restrict__ asrc,
                                               const float* __restrict__ adst,
                                               unsigned* __restrict__ gkey, int total) {
    __shared__ float red[256];
    const int tid = threadIdx.x;
    float m = -3.402823466e+38f;
    for (int i = blockIdx.x * blockDim.x + tid; i < total; i += gridDim.x * blockDim.x) {
        const int e  = i >> 2;
        const int hd = i & 3;
        float att = asrc[src[e] * N_HEADS + hd] + adst[dst[e] * N_HEADS + hd];
        att = (att >= 0.f) ? att : NEG_SLOPE * att;
        m = fmaxf(m, att);
    }
    red[tid] = m;
    __syncthreads();
    for (int s = 128; s > 0; s >>= 1) {
        if (tid < s) red[tid] = fmaxf(red[tid], red[tid + s]);
        __syncthreads();
    }
    if (tid == 0) atomicMax(gkey, fkey(red[0]));
}

// ---------- 4) exp + segment-sum of normalizers ----------
__global__ __launch_bounds__(256) void gat_sum(const int* __restrict__ src,
                                               const int* __restrict__ dst,
                                               const float* __restrict__ asrc,
                                               const float* __restrict__ adst,
                                               const unsigned* __restrict__ gkey,
                                               float* __restrict__ att_sum, int total) {
    int i = blockIdx.x * blockDim.x + threadIdx.x;
    if (i >= total) return;
    const int e  = i >> 2;
    const int hd = i & 3;
    const int d  = dst[e];
    const float gm = funkey(*gkey);
    float att = asrc[src[e] * N_HEADS + hd] + adst[d * N_HEADS + hd];
    att = (att >= 0.f) ? att : NEG_SLOPE * att;
    atomicAdd(&att_sum[d * N_HEADS + hd], __expf(att - gm));
}

// ---------- 5) weighted scatter: one wave per edge, 4 features per lane ----------
__global__ __launch_bounds__(256) void gat_scatter(const int* __restrict__ src,
                                                   const int* __restrict__ dst,
                                                   const float* __restrict__ asrc,
                                                   const float* __restrict__ adst,
                                                   const float* __restrict__ att_sum,
                                                   const unsigned* __restrict__ gkey,
                                                   const float* __restrict__ hbuf,
                                                   float* __restrict__ out, int n_edges) {
    const int gid = blockIdx.x * blockDim.x + threadIdx.x;
    const int e   = gid >> 5;
    if (e >= n_edges) return;
    const int lane = threadIdx.x & 31;
    const int s = src[e];
    const int d = dst[e];
    const int hd = lane >> 3;                  // 8 lanes per head (32 feats / 4 per lane)
    const float gm = funkey(*gkey);
    float att = asrc[s * N_HEADS + hd] + adst[d * N_HEADS + hd];
    att = (att >= 0.f) ? att : NEG_SLOPE * att;
    const float w = __expf(att - gm) / (att_sum[d * N_HEADS + hd] + GAT_EPS);

    const float4 hv = *reinterpret_cast<const float4*>(hbuf + (size_t)s * N_OUT_C + lane * 4);
    float* op = out + (size_t)d * N_OUT_C + lane * 4;
    atomicAdd(op + 0, w * hv.x);
    atomicAdd(op + 1, w * hv.y);
    atomicAdd(op + 2, w * hv.z);
    atomicAdd(op + 3, w * hv.w);
}

extern "C" void kernel_launch(void* const* d_in, const int* in_sizes, int n_in,
                              void* d_out, int out_size, void* d_ws, size_t ws_size,
                              hipStream_t stream) {
    const float* x  = (const float*)d_in[0];   // (N, 128)
    const float* W  = (const float*)d_in[1];   // (128, 128)
    const float* a  = (const float*)d_in[2];   // (4, 64)
    const int*  ei  = (const int*)d_in[3];     // (2, E)

    const int n_nodes = in_sizes[0] / N_IN_F;
    const int n_edges = in_sizes[3] / 2;
    const int* src = ei;
    const int* dst = ei + n_edges;

    float* out = (float*)d_out;

    // workspace layout (floats)
    float* ws = (float*)d_ws;
    float*    hbuf      = ws;                                   // N*128
    float*    alpha_src = hbuf + (size_t)n_nodes * N_OUT_C;     // N*4
    float*    alpha_dst = alpha_src + (size_t)n_nodes * N_HEADS;
    float*    att_sum   = alpha_dst + (size_t)n_nodes * N_HEADS;
    unsigned* gkey      = (unsigned*)(att_sum + (size_t)n_nodes * N_HEADS);

    const int nout = n_nodes * N_OUT_C;
    const int nsum = n_nodes * N_HEADS;
    const int totEH = n_edges * N_HEADS;

    // 0) zero out / att_sum / gmax
    {
        int blocks = (nout + 255) / 256;
        gat_zero<<<blocks, 256, 0, stream>>>(out, att_sum, gkey, nout, nsum);
    }
    // 1) GEMM via fp32 WMMA
    {
        int blocks = (n_nodes + 15) / 16;
        gat_gemm_wmma<<<blocks, 256, 0, stream>>>(x, W, hbuf, n_nodes);
    }
    // 2) per-node alphas
    {
        int blocks = (nsum + 255) / 256;
        gat_alpha<<<blocks, 256, 0, stream>>>(hbuf, a, alpha_src, alpha_dst, n_nodes);
    }
    // 3) global max
    {
        gat_max<<<2048, 256, 0, stream>>>(src, dst, alpha_src, alpha_dst, gkey, totEH);
    }
    // 4) normalizer segment-sum
    {
        int blocks = (totEH + 255) / 256;
        gat_sum<<<blocks, 256, 0, stream>>>(src, dst, alpha_src, alpha_dst, gkey, att_sum, totEH);
    }
    // 5) weighted scatter (wave per edge)
    {
        long long threads = (long long)n_edges * 32;
        int blocks = (int)((threads + 255) / 256);
        gat_scatter<<<blocks, 256, 0, stream>>>(src, dst, alpha_src, alpha_dst,
                                                att_sum, gkey, hbuf, out, n_edges);
    }
}